// CapsuleNetwork_29557964931532
// MI455X (gfx1250) — compile-verified
//
#include <hip/hip_runtime.h>
#include <math.h>

typedef __attribute__((ext_vector_type(2))) float v2f;
typedef __attribute__((ext_vector_type(8))) float v8f;

#define NUM_CAPS 8
#define CAPS_DIM 8
#define OUT_CAPS 2
#define OUT_DIM 16
#define SEQ_L 2048

static __device__ __forceinline__ float fast_rcp(float a) {
    return __builtin_amdgcn_rcpf(a);
}
static __device__ __forceinline__ float fast_sqrt(float a) {
    return __builtin_amdgcn_sqrtf(a);
}

// One block per batch row. 128 threads = 4 waves.
// Wave p handles capsules 2p (lanes 0-15) and 2p+1 (lanes 16-31) = channels 16p..16p+15.
__global__ __launch_bounds__(128)
void capsnet_fused_kernel(const float* __restrict__ x,
                          const float* __restrict__ conv_w,
                          const float* __restrict__ conv_b,
                          const float* __restrict__ attn_w,
                          const float* __restrict__ attn_b,
                          const float* __restrict__ W_digit,
                          float* __restrict__ out, int B)
{
    __shared__ float xl[SEQ_L + 4];                  // zero-padded input row
    __shared__ float aggL[NUM_CAPS * CAPS_DIM];      // agg[b, n, d]

    const int b    = blockIdx.x;
    const int tid  = threadIdx.x;
    const int wave = tid >> 5;
    const int lane = tid & 31;
    const int half = lane >> 4;     // 0 -> capsule 2p, 1 -> capsule 2p+1
    const int t    = lane & 15;     // position-in-tile (B/C col) or channel row (A)

    // ---------------- stage 0: x row -> LDS, zero padded ----------------
    for (int i = tid; i < SEQ_L + 2; i += 128) {
        float v = 0.0f;
        if (i >= 1 && i <= SEQ_L) v = x[(size_t)b * SEQ_L + (i - 1)];
        xl[i] = v;
    }
    __syncthreads();

    // ---------------- stage 1: conv(WMMA f32 16x16x4) + squash + online softmax ----
    // A matrix: 16 channels x 4 taps; 4th tap carries the conv bias (B row3 = 1.0).
    // ISA layout (32-bit A 16x4): lanes 0-15: VGPR0=K0, VGPR1=K1 ; lanes 16-31: VGPR0=K2, VGPR1=K3
    const int ch = wave * 16 + t;
    v2f Amat;
    if (half == 0) { Amat.x = conv_w[ch * 3 + 0]; Amat.y = conv_w[ch * 3 + 1]; }
    else           { Amat.x = conv_w[ch * 3 + 2]; Amat.y = conv_b[ch];         }

    float awv[CAPS_DIM];
#pragma unroll
    for (int d = 0; d < CAPS_DIM; ++d) awv[d] = attn_w[d];
    const float ab = attn_b[0];

    // online-softmax state for this lane's position subset of its capsule
    float mrun = -INFINITY;
    float zrun = 0.0f;
    float S[CAPS_DIM];
#pragma unroll
    for (int d = 0; d < CAPS_DIM; ++d) S[d] = 0.0f;

#pragma unroll 2
    for (int L0 = 0; L0 < SEQ_L; L0 += 16) {
        // B matrix 4x16: row K = tap K; x[p+k-1] with p = L0+t; xl index shifted +1.
        // Layout: VGPR0 = {K0 (lanes0-15), K2 (lanes16-31)}, VGPR1 = {K1, K3}.
        v2f Bmat;
        if (half == 0) { Bmat.x = xl[L0 + t];     Bmat.y = xl[L0 + t + 1]; }
        else           { Bmat.x = xl[L0 + t + 2]; Bmat.y = 1.0f;           }

        v8f cz = {};
        v8f h = __builtin_amdgcn_wmma_f32_16x16x4_f32(
            false, Amat, false, Bmat, (short)0, cz, false, false);
        // per-lane: h[0..7] = 8 capsule dims (incl. bias) at position L0+t, capsule 2*wave+half

        float sq = 0.0f;
#pragma unroll
        for (int d = 0; d < CAPS_DIM; ++d) sq += h[d] * h[d];
        float nrm = fast_sqrt(sq);
        // squash factor: sq / ((1+sq)(nrm+1e-8)) via fast reciprocal (1-ulp ok here)
        float f = sq * fast_rcp((1.0f + sq) * (nrm + 1e-8f));

        float caps[CAPS_DIM];
        float logit = ab;
#pragma unroll
        for (int d = 0; d < CAPS_DIM; ++d) {
            caps[d] = h[d] * f;
            logit += caps[d] * awv[d];
        }

        // online softmax over L (flash-style)
        float nm = fmaxf(mrun, logit);
        float sc = __expf(mrun - nm);
        float e  = __expf(logit - nm);
        zrun = zrun * sc + e;
#pragma unroll
        for (int d = 0; d < CAPS_DIM; ++d) S[d] = S[d] * sc + e * caps[d];
        mrun = nm;
    }

    // merge the 16 lanes of each capsule half (xor network stays within halves)
#pragma unroll
    for (int off = 1; off < 16; off <<= 1) {
        float m2 = __shfl_xor(mrun, off, 32);
        float z2 = __shfl_xor(zrun, off, 32);
        float S2[CAPS_DIM];
#pragma unroll
        for (int d = 0; d < CAPS_DIM; ++d) S2[d] = __shfl_xor(S[d], off, 32);
        float M  = fmaxf(mrun, m2);
        float a1 = __expf(mrun - M);
        float a2 = __expf(m2 - M);
        zrun = zrun * a1 + z2 * a2;
#pragma unroll
        for (int d = 0; d < CAPS_DIM; ++d) S[d] = S[d] * a1 + S2[d] * a2;
        mrun = M;
    }

    if (t == 0) {
        int cap = wave * 2 + half;
        float inv = fast_rcp(zrun);
#pragma unroll
        for (int d = 0; d < CAPS_DIM; ++d) aggL[cap * CAPS_DIM + d] = S[d] * inv;
    }
    __syncthreads();

    // ---------------- stage 2: u_hat + dynamic routing (wave 0 only) ----------------
    if (wave == 0) {
        const int o = lane >> 4;      // output capsule 0/1
        const int j = lane & 15;      // output dim

        // u_hat[i][o][j] per lane, i = 0..7
        float u[NUM_CAPS];
#pragma unroll
        for (int i = 0; i < NUM_CAPS; ++i) {
            float acc = 0.0f;
            const float* Wrow = W_digit + ((i * OUT_CAPS + o) * OUT_DIM + j) * CAPS_DIM;
#pragma unroll
            for (int d = 0; d < CAPS_DIM; ++d)
                acc += Wrow[d] * aggL[i * CAPS_DIM + d];
            u[i] = acc;
        }

        float blog[NUM_CAPS];
#pragma unroll
        for (int i = 0; i < NUM_CAPS; ++i) blog[i] = 0.0f;

        float vj = 0.0f, sqv = 0.0f;
#pragma unroll
        for (int it = 0; it < 3; ++it) {
            // c = softmax over o of blog; s[o][j] = sum_i c[i][o] * u_hat[i][o][j]
            float s = 0.0f;
#pragma unroll
            for (int i = 0; i < NUM_CAPS; ++i) {
                float bo = blog[i];
                float bq = __shfl_xor(bo, 16, 32);       // partner output capsule
                float mx = fmaxf(bo, bq);
                float e0 = __expf(bo - mx);
                float e1 = __expf(bq - mx);
                s += e0 * fast_rcp(e0 + e1) * u[i];
            }
            // v = squash(s) along j (16 lanes of this o-half)
            float sq = s * s;
#pragma unroll
            for (int off = 1; off < 16; off <<= 1) sq += __shfl_xor(sq, off, 32);
            float nrm = fast_sqrt(sq);
            float f = sq * fast_rcp((1.0f + sq) * (nrm + 1e-8f));
            vj = s * f;

            if (it != 2) {
#pragma unroll
                for (int i = 0; i < NUM_CAPS; ++i) {
                    float tt = u[i] * vj;
#pragma unroll
                    for (int off = 1; off < 16; off <<= 1) tt += __shfl_xor(tt, off, 32);
                    blog[i] += tt;
                }
            } else {
                sqv = vj * vj;
#pragma unroll
                for (int off = 1; off < 16; off <<= 1) sqv += __shfl_xor(sqv, off, 32);
            }
        }

        // digit_out: out[B*2 + b*32 + o*16 + j]
        out[(size_t)B * OUT_CAPS + (size_t)b * (OUT_CAPS * OUT_DIM) + lane] = vj;

        // probs = softmax over o of ||v||
        float len = fast_sqrt(sqv);
        float lq  = __shfl_xor(len, 16, 32);
        float mx  = fmaxf(len, lq);
        float e0  = __expf(len - mx);
        float e1  = __expf(lq - mx);
        if (j == 0) out[(size_t)b * OUT_CAPS + o] = e0 * fast_rcp(e0 + e1);
    }
}

extern "C" void kernel_launch(void* const* d_in, const int* in_sizes, int n_in,
                              void* d_out, int out_size, void* d_ws, size_t ws_size,
                              hipStream_t stream) {
    (void)n_in; (void)out_size; (void)d_ws; (void)ws_size;
    const float* x       = (const float*)d_in[0];
    const float* conv_w  = (const float*)d_in[1];
    const float* conv_b  = (const float*)d_in[2];
    const float* attn_w  = (const float*)d_in[3];
    const float* attn_b  = (const float*)d_in[4];
    const float* W_digit = (const float*)d_in[5];
    const int B = in_sizes[0] / SEQ_L;   // 2048

    capsnet_fused_kernel<<<B, 128, 0, stream>>>(
        x, conv_w, conv_b, attn_w, attn_b, W_digit, (float*)d_out, B);
}